// GCN_22849226015225
// MI455X (gfx1250) — compile-verified
//
#include <hip/hip_runtime.h>

// GCN 2-layer forward for MI455X (gfx1250, wave32).
// Pipeline: degrees -> norm -> WMMA GEMM1 -> edge scatter -> bias/relu
//           -> WMMA GEMM2 -> edge scatter -> bias -> out.
// GEMMs use V_WMMA_F32_16X16X4_F32 (fp32 exact). Scatter is the bandwidth
// bound (~2.5 GB @ 23.3 TB/s); laid out so consecutive lanes touch
// consecutive features of one edge (coalesced gather + coalesced f32 atomics).

#define IN_F  256
#define HID_F 128
#define OUT_F 64

typedef __attribute__((ext_vector_type(2))) float v2f;
typedef __attribute__((ext_vector_type(8))) float v8f;

// ---------------------------------------------------------------- utilities
__global__ void zero_f32(float* __restrict__ p, long long n) {
  long long i = (long long)blockIdx.x * blockDim.x + threadIdx.x;
  long long stride = (long long)gridDim.x * blockDim.x;
  for (; i < n; i += stride) p[i] = 0.0f;
}

__global__ void degree_count(const int* __restrict__ src, const int* __restrict__ dst,
                             int* __restrict__ degO, int* __restrict__ degI, int E) {
  int i = blockIdx.x * blockDim.x + threadIdx.x;
  if (i < E) {
    atomicAdd(&degO[src[i]], 1);
    atomicAdd(&degI[dst[i]], 1);
  }
}

__global__ void make_norm(const int* __restrict__ degO, const int* __restrict__ degI,
                          float* __restrict__ nO, float* __restrict__ nI, int n) {
  int i = blockIdx.x * blockDim.x + threadIdx.x;
  if (i < n) {
    int dO = degO[i]; if (dO < 1) dO = 1;
    int dI = degI[i]; if (dI < 1) dI = 1;
    nO[i] = 1.0f / sqrtf((float)dO);
    nI[i] = 1.0f / sqrtf((float)dI);
  }
}

// ------------------------------------------------- WMMA GEMM, row-scaled A
// Y[M x NC] = diag(rs) * X[M x K] @ W[K x NC]
// One wave per 16x16 output tile; K stepped by 4 via V_WMMA_F32_16X16X4_F32.
// A layout (32b 16x4):  lane%16 = row M; VGPR0 holds K = 2*(lane/16),
//                       VGPR1 holds K = 2*(lane/16)+1.
// B layout (32b 4x16):  lane%16 = col N; same K-split across lane halves.
// C/D layout (32b 16x16): lane%16 = col N; VGPR r -> row r + 8*(lane/16).
template <int K, int NC>
__global__ __launch_bounds__(256) void gemm_rowscale_wmma(
    const float* __restrict__ X, const float* __restrict__ W,
    const float* __restrict__ rs, float* __restrict__ Y, int M) {
  const int lane = threadIdx.x & 31;
  const int wave = (int)(blockIdx.x * (blockDim.x >> 5) + (threadIdx.x >> 5));
  constexpr int tilesN = NC / 16;
  const int tileM = wave / tilesN;
  const int tileN = wave % tilesN;
  if (tileM * 16 >= M) return;           // wave-uniform: EXEC stays all-1s

  const int lrow = lane & 15;            // A-row / B-col / D-col
  const int hh   = lane >> 4;            // lane half selects K pair
  int m = tileM * 16 + lrow;
  if (m >= M) m = M - 1;                 // defensive clamp (address-only)
  const float s = rs[m];
  const float* __restrict__ xrow = X + (long long)m * K;
  const float* __restrict__ wcol = W + tileN * 16 + lrow;

  v8f acc = {0.f, 0.f, 0.f, 0.f, 0.f, 0.f, 0.f, 0.f};
#pragma unroll 8
  for (int k = 0; k < K; k += 4) {
    v2f a, b;
    a.x = xrow[k + 2 * hh] * s;
    a.y = xrow[k + 2 * hh + 1] * s;
    b.x = wcol[(long long)(k + 2 * hh) * NC];
    b.y = wcol[(long long)(k + 2 * hh + 1) * NC];
    acc = __builtin_amdgcn_wmma_f32_16x16x4_f32(
        /*neg_a=*/false, a, /*neg_b=*/false, b,
        /*c_mod=*/(short)0, acc, /*reuse_a=*/false, /*reuse_b=*/false);
  }

  float* __restrict__ yout = Y + (long long)(tileM * 16) * NC + tileN * 16 + lrow;
#pragma unroll
  for (int r = 0; r < 8; ++r) {
    int row = r + 8 * hh;
    if (tileM * 16 + row < M)
      yout[(long long)row * NC] = acc[r];
  }
}

// --------------------------------------------------------- edge scatter-sum
// agg[dst[e], f] += Y[src[e], f]; consecutive threads cover consecutive f of
// one edge -> coalesced 128B gathers and coalesced global_atomic_add_f32.
template <int F>
__global__ void scatter_add(const float* __restrict__ Y, const int* __restrict__ src,
                            const int* __restrict__ dst, float* __restrict__ agg, int E) {
  long long i = (long long)blockIdx.x * blockDim.x + threadIdx.x;
  const long long total = (long long)E * F;
  const long long stride = (long long)gridDim.x * blockDim.x;
  for (; i < total; i += stride) {
    int e = (int)(i / F);
    int f = (int)(i - (long long)e * F);
    int s = src[e];
    int d = dst[e];
    float v = Y[(long long)s * F + f];
    atomicAdd(&agg[(long long)d * F + f], v);
  }
}

// -------------------------------------------------------- bias + norm (+relu)
template <int F, bool RELU>
__global__ void finish_bias(const float* __restrict__ agg, const float* __restrict__ nI,
                            const float* __restrict__ bias, float* __restrict__ out,
                            long long total) {
  long long i = (long long)blockIdx.x * blockDim.x + threadIdx.x;
  const long long stride = (long long)gridDim.x * blockDim.x;
  for (; i < total; i += stride) {
    int node = (int)(i / F);
    int f = (int)(i - (long long)node * F);
    float v = agg[i] * nI[node] + bias[f];
    out[i] = RELU ? fmaxf(v, 0.0f) : v;
  }
}

// ---------------------------------------------------------------- launcher
extern "C" void kernel_launch(void* const* d_in, const int* in_sizes, int n_in,
                              void* d_out, int out_size, void* d_ws, size_t ws_size,
                              hipStream_t stream) {
  const float* feat = (const float*)d_in[0];
  const int*   src  = (const int*)d_in[1];
  const int*   dst  = (const int*)d_in[2];
  const float* W1   = (const float*)d_in[3];
  const float* b1   = (const float*)d_in[4];
  const float* W2   = (const float*)d_in[5];
  const float* b2   = (const float*)d_in[6];
  float* out = (float*)d_out;

  const int N = in_sizes[0] / IN_F;   // 100000
  const int E = in_sizes[1];          // 1600000

  // workspace carve-out (256B aligned slices)
  char* ws = (char*)d_ws;
  size_t off = 0;
  auto carve = [&](size_t bytes) -> void* {
    void* p = ws + off;
    off += (bytes + 255) & ~(size_t)255;
    return p;
  };
  int*   degBoth = (int*)carve((size_t)2 * N * sizeof(int));  // degO | degI
  int*   degO = degBoth;
  int*   degI = degBoth + N;
  float* nO   = (float*)carve((size_t)N * sizeof(float));
  float* nI   = (float*)carve((size_t)N * sizeof(float));
  float* Y1   = (float*)carve((size_t)N * HID_F * sizeof(float));
  float* AG1  = (float*)carve((size_t)N * HID_F * sizeof(float));
  // Y1 is dead after scatter-1: reuse its space for layer-2 buffers
  float* Y2  = Y1;                       // N*OUT_F
  float* AG2 = Y1 + (size_t)N * OUT_F;   // N*OUT_F
  (void)ws_size; (void)n_in; (void)out_size;

  const int TB = 256;
  auto blocks1d = [](long long n, int tb) {
    long long b = (n + tb - 1) / tb;
    return (unsigned)(b > 1048576 ? 1048576 : b);
  };

  // 1) degrees + normalization
  zero_f32<<<blocks1d((long long)2 * N, TB), TB, 0, stream>>>((float*)degBoth, (long long)2 * N);
  degree_count<<<(E + TB - 1) / TB, TB, 0, stream>>>(src, dst, degO, degI, E);
  make_norm<<<(N + TB - 1) / TB, TB, 0, stream>>>(degO, degI, nO, nI, N);

  // 2) layer 1: GEMM -> scatter -> bias/relu (in place over AG1)
  zero_f32<<<blocks1d((long long)N * HID_F, TB), TB, 0, stream>>>(AG1, (long long)N * HID_F);
  {
    int tiles = ((N + 15) / 16) * (HID_F / 16);
    int blocks = (tiles * 32 + TB - 1) / TB;
    gemm_rowscale_wmma<IN_F, HID_F><<<blocks, TB, 0, stream>>>(feat, W1, nO, Y1, N);
  }
  scatter_add<HID_F><<<blocks1d((long long)E * HID_F, TB), TB, 0, stream>>>(Y1, src, dst, AG1, E);
  finish_bias<HID_F, true><<<blocks1d((long long)N * HID_F, TB), TB, 0, stream>>>(
      AG1, nI, b1, AG1, (long long)N * HID_F);

  // 3) layer 2: GEMM -> scatter -> bias
  zero_f32<<<blocks1d((long long)N * OUT_F, TB), TB, 0, stream>>>(AG2, (long long)N * OUT_F);
  {
    int tiles = ((N + 15) / 16) * (OUT_F / 16);
    int blocks = (tiles * 32 + TB - 1) / TB;
    gemm_rowscale_wmma<HID_F, OUT_F><<<blocks, TB, 0, stream>>>(AG1, W2, nO, Y2, N);
  }
  scatter_add<OUT_F><<<blocks1d((long long)E * OUT_F, TB), TB, 0, stream>>>(Y2, src, dst, AG2, E);
  finish_bias<OUT_F, false><<<blocks1d((long long)N * OUT_F, TB), TB, 0, stream>>>(
      AG2, nI, b2, out, (long long)N * OUT_F);
}